// ImaginariumModelBaseLine_69260642615608
// MI455X (gfx1250) — compile-verified
//
#include <hip/hip_runtime.h>

// Problem constants (from reference): B=4096, C=36, W=50, D=512, fp32.
#define C_ 36
#define W_ 50
#define D_ 512
#define EPSF 1e-8f

#define TSTRIDE 516   // padded LDS row stride (floats): 516 % 64 rotates banks
#define TROWS   64    // 50 word rows padded to 64 (4 WMMA tiles of 16)

typedef __attribute__((ext_vector_type(2))) float v2f;
typedef __attribute__((ext_vector_type(8))) float v8f;

__global__ __launch_bounds__(256)
void imaginarium_fused_kernel(const float* __restrict__ img,   // [B,C,D]
                              const float* __restrict__ txt,   // [B,W,D]
                              const float* __restrict__ y,     // [B,C]
                              float* __restrict__ out)         // [B,C]
{
    __shared__ __attribute__((aligned(16))) float sTxt[TROWS * TSTRIDE]; // 132 KB
    __shared__ __attribute__((aligned(16))) float sChosen[D_];
    __shared__ __attribute__((aligned(16))) float sFinal[D_];
    __shared__ float sNtxt[TROWS];
    __shared__ float sKpart[2][TROWS];
    __shared__ float sWeights[TROWS];
    __shared__ float sDots[C_ + 4];
    __shared__ float sNimg[C_ + 4];

    const int b    = blockIdx.x;
    const int tid  = threadIdx.x;
    const int lane = tid & 31;
    const int wave = tid >> 5;

    // ---- Phase 0: argmax over y[b, 0..C) (first-max semantics, all threads redundantly,
    //      uniform addresses -> scalarized loads) ----
    const float* yb = y + (size_t)b * C_;
    float mbest = yb[0];
    int   idx   = 0;
    for (int c = 1; c < C_; ++c) {
        float v = yb[c];
        if (v > mbest) { mbest = v; idx = c; }
    }

    // ---- Phase 1: stage txt[b] -> LDS (float4, coalesced), zero pad rows,
    //      load chosen = img[b, idx, :] -> LDS ----
    {
        const float4* tsrc = (const float4*)(txt + (size_t)b * W_ * D_);
        for (int i = tid; i < (W_ * D_) / 4; i += 256) {
            float4 v = tsrc[i];
            int row = i >> 7;            // i / (D_/4)
            int col = (i & 127) << 2;    // (i % 128) * 4
            *(float4*)&sTxt[row * TSTRIDE + col] = v;
        }
        for (int i = tid; i < (TROWS - W_) * TSTRIDE; i += 256)
            sTxt[W_ * TSTRIDE + i] = 0.0f;
        const float4* csrc = (const float4*)(img + ((size_t)b * C_ + idx) * D_);
        if (tid < D_ / 4)
            *(float4*)&sChosen[tid * 4] = csrc[tid];
    }
    __syncthreads();

    // ---- Phase 2 (all waves): ||chosen||^2 (wave 0), txt row norms, and
    //      WMMA word-dots: dot(chosen, txt_w) via V_WMMA_F32_16X16X4_F32 ----
    float nch2 = 0.0f;  // valid in wave 0 only
    if (wave == 0) {
        float s = 0.0f;
        for (int q = 0; q < D_ / 32; ++q) {
            float v = sChosen[lane + 32 * q];
            s += v * v;
        }
        for (int mk = 16; mk; mk >>= 1) s += __shfl_xor(s, mk, 32);
        nch2 = s;
    }
    for (int w = wave; w < W_; w += 8) {
        const float* rp = &sTxt[w * TSTRIDE];
        float s = 0.0f;
        for (int q = 0; q < D_ / 32; ++q) {
            float v = rp[lane + 32 * q];
            s += v * v;
        }
        for (int mk = 16; mk; mk >>= 1) s += __shfl_xor(s, mk, 32);
        if (lane == 0) sNtxt[w] = s;
    }
    {
        // wave j -> tile (j>>1), k-half (j&1). Tile rows M0..M0+15, K range kh*256..+255.
        const int tile = wave >> 1;
        const int kh   = wave & 1;
        const int M0   = tile * 16;
        // A layout (16x4 f32): VGPR0 = A[m, K0 | K0+2], VGPR1 = A[m, K0+1 | K0+3]
        //   lane 0..15 -> k pair (K0, K0+1), lane 16..31 -> (K0+2, K0+3), row = M0 + (lane&15)
        // B (4x16 f32, all columns = chosen chunk): same k-pair selection, broadcast address.
        const float* arow = &sTxt[(M0 + (lane & 15)) * TSTRIDE];
        const int kbase = kh * 256 + ((lane >> 4) << 1);
        v8f acc = {0.f, 0.f, 0.f, 0.f, 0.f, 0.f, 0.f, 0.f};
        for (int k = 0; k < 256; k += 4) {
            const int ko = kbase + k;
            v2f a  = *(const v2f*)&arow[ko];
            v2f bb = *(const v2f*)&sChosen[ko];
            acc = __builtin_amdgcn_wmma_f32_16x16x4_f32(
                false, a, false, bb, (short)0, acc, false, false);
        }
        // D layout: VGPR r = rows {M0+r (lanes 0-15), M0+8+r (lanes 16-31)}; all columns equal.
        if (lane == 0) {
            for (int r = 0; r < 8; ++r) sKpart[kh][M0 + r] = acc[r];
        } else if (lane == 16) {
            for (int r = 0; r < 8; ++r) sKpart[kh][M0 + 8 + r] = acc[r];
        }
    }
    __syncthreads();

    // ---- Phase 3 (wave 0): word logits = dot/(max(|a|,eps)*max(|b|,eps)), softmax over W ----
    if (wave == 0) {
        const float na = fmaxf(__builtin_sqrtf(nch2), EPSF);
        const int w0 = lane, w1 = lane + 32;
        float l0 = -__builtin_inff(), l1 = -__builtin_inff();
        {
            float d  = sKpart[0][w0] + sKpart[1][w0];
            float nb = fmaxf(__builtin_sqrtf(sNtxt[w0]), EPSF);
            l0 = d / (na * nb);
        }
        if (w1 < W_) {
            float d  = sKpart[0][w1] + sKpart[1][w1];
            float nb = fmaxf(__builtin_sqrtf(sNtxt[w1]), EPSF);
            l1 = d / (na * nb);
        }
        float mx = fmaxf(l0, l1);
        for (int mk = 16; mk; mk >>= 1) mx = fmaxf(mx, __shfl_xor(mx, mk, 32));
        float e0 = __builtin_expf(l0 - mx);
        float e1 = (w1 < W_) ? __builtin_expf(l1 - mx) : 0.0f;
        float s = e0 + e1;
        for (int mk = 16; mk; mk >>= 1) s += __shfl_xor(s, mk, 32);
        float inv = 1.0f / s;
        sWeights[w0] = e0 * inv;
        if (w1 < W_) sWeights[w1] = e1 * inv;
    }
    __syncthreads();

    // ---- Phase 4: final[d] = sum_w weights[w] * txt[w,d]  (column access, conflict-free) ----
    for (int d = tid; d < D_; d += 256) {
        float acc = 0.0f;
        for (int w = 0; w < W_; ++w)
            acc = fmaf(sWeights[w], sTxt[w * TSTRIDE + d], acc);
        sFinal[d] = acc;
    }
    __syncthreads();

    // ---- Phase 5 (all waves): single coalesced global pass over img[b]:
    //      fused dot(final, img_c) and ||img_c||^2 ----
    {
        const float4* ib = (const float4*)(img + (size_t)b * C_ * D_);
        for (int c = wave; c < C_; c += 8) {
            const float4* rp = ib + c * (D_ / 4);
            float dot = 0.0f, nsq = 0.0f;
            for (int q = 0; q < 4; ++q) {
                const int i = lane + 32 * q;
                float4 v = rp[i];
                float4 f = *(const float4*)&sFinal[i * 4];
                dot += v.x * f.x + v.y * f.y + v.z * f.z + v.w * f.w;
                nsq += v.x * v.x + v.y * v.y + v.z * v.z + v.w * v.w;
            }
            for (int mk = 16; mk; mk >>= 1) {
                dot += __shfl_xor(dot, mk, 32);
                nsq += __shfl_xor(nsq, mk, 32);
            }
            if (lane == 0) { sDots[c] = dot; sNimg[c] = nsq; }
        }
    }
    __syncthreads();

    // ---- Phase 6 (wave 0): ||final||, card logits, softmax over C, store ----
    if (wave == 0) {
        float s = 0.0f;
        for (int q = 0; q < D_ / 32; ++q) {
            float v = sFinal[lane + 32 * q];
            s += v * v;
        }
        for (int mk = 16; mk; mk >>= 1) s += __shfl_xor(s, mk, 32);
        const float nf = fmaxf(__builtin_sqrtf(s), EPSF);

        const int c0 = lane, c1 = lane + 32;
        float l0 = -__builtin_inff(), l1 = -__builtin_inff();
        if (c0 < C_) l0 = sDots[c0] / (nf * fmaxf(__builtin_sqrtf(sNimg[c0]), EPSF));
        if (c1 < C_) l1 = sDots[c1] / (nf * fmaxf(__builtin_sqrtf(sNimg[c1]), EPSF));
        float mx = fmaxf(l0, l1);
        for (int mk = 16; mk; mk >>= 1) mx = fmaxf(mx, __shfl_xor(mx, mk, 32));
        float e0 = (c0 < C_) ? __builtin_expf(l0 - mx) : 0.0f;
        float e1 = (c1 < C_) ? __builtin_expf(l1 - mx) : 0.0f;
        float ssum = e0 + e1;
        for (int mk = 16; mk; mk >>= 1) ssum += __shfl_xor(ssum, mk, 32);
        const float inv = 1.0f / ssum;
        float* ob = out + (size_t)b * C_;
        if (c0 < C_) ob[c0] = e0 * inv;
        if (c1 < C_) ob[c1] = e1 * inv;
    }
}

extern "C" void kernel_launch(void* const* d_in, const int* in_sizes, int n_in,
                              void* d_out, int out_size, void* d_ws, size_t ws_size,
                              hipStream_t stream) {
    (void)n_in; (void)d_ws; (void)ws_size; (void)out_size;
    const float* img = (const float*)d_in[0];  // [B, C, D] fp32
    const float* txt = (const float*)d_in[1];  // [B, W, D] fp32
    const float* yv  = (const float*)d_in[2];  // [B, C]    fp32
    float* out = (float*)d_out;                // [B, C]    fp32
    const int B = in_sizes[2] / C_;
    imaginarium_fused_kernel<<<dim3(B), dim3(256), 0, stream>>>(img, txt, yv, out);
}